// CascadeClassifierGNN_16939351015795
// MI455X (gfx1250) — compile-verified
//
#include <hip/hip_runtime.h>

typedef float v2f __attribute__((ext_vector_type(2)));
typedef float v8f __attribute__((ext_vector_type(8)));

#define EPSBN 1e-5f

// ---------------- utility kernels (32-bit indexing; all sizes < 2^31) ----------------

__global__ void k_fill(float* __restrict__ p, float val, unsigned nElem) {
  unsigned i = blockIdx.x * blockDim.x + threadIdx.x;
  if (i < nElem) p[i] = val;
}

__global__ void k_edge_deg(const int* __restrict__ dst, float* __restrict__ deg, unsigned E) {
  unsigned e = blockIdx.x * blockDim.x + threadIdx.x;
  if (e < E) atomicAdd(&deg[dst[e]], 1.0f);
}

__global__ void k_rsqrt_inplace(float* __restrict__ p, unsigned n) {
  unsigned i = blockIdx.x * blockDim.x + threadIdx.x;
  if (i < n) p[i] = rsqrtf(p[i]);
}

// zero-pad x[N,K] -> xp[N,KPAD] so the GEMM needs no K-guards
__global__ void k_pad(const float* __restrict__ x, float* __restrict__ xp,
                      unsigned n, int K, int KPAD) {
  unsigned idx = blockIdx.x * blockDim.x + threadIdx.x;
  if (idx >= n * (unsigned)KPAD) return;
  unsigned i = idx / (unsigned)KPAD;
  int j = (int)(idx - i * (unsigned)KPAD);
  xp[idx] = (j < K) ? x[i * (unsigned)K + j] : 0.0f;
}

// ---------------- WMMA GEMM: out[N,64] = A[N,KPAD] @ W[K,64] (f32, exact) ----------------
// 256 threads = 8 waves; each wave computes a 16x64 tile with V_WMMA_F32_16X16X4_F32.
// W staged in LDS pre-swizzled as (K,K+1) float2 pairs -> B fragments are ds_load_b64.
// A rows pre-padded to KPAD -> A fragments are unconditional global_load_b64.
// Full tiles take a scalar-branch fast path with immediate-offset store clauses.

template <int KPAD>
__global__ __launch_bounds__(256) void k_gemm_wmma(
    const float* __restrict__ A, const float* __restrict__ W,
    float* __restrict__ out, int n, int K) {
  __shared__ v2f sW2[(KPAD / 2) * 64];

  int tid = threadIdx.x;
  for (int idx = tid; idx < (KPAD / 2) * 64; idx += 256) {
    int p = idx >> 6;         // K-pair index
    int col = idx & 63;
    v2f val;
    val.x = (2 * p     < K) ? W[(2 * p    ) * 64 + col] : 0.0f;
    val.y = (2 * p + 1 < K) ? W[(2 * p + 1) * 64 + col] : 0.0f;
    sW2[idx] = val;
  }
  __syncthreads();

  int wave = tid >> 5;
  int lane = tid & 31;
  int rowBase = (int)blockIdx.x * 128 + wave * 16;   // wave-uniform
  int m    = lane & 15;      // M (for A) / N (for B) index within tile
  int half = lane >> 4;      // lanes 0-15 -> K+0,K+1 ; lanes 16-31 -> K+2,K+3

  int arow = rowBase + m;
  if (arow > n - 1) arow = n - 1;      // clamp OOB rows (stores are guarded)
  const float* Arow = A + (long long)arow * KPAD;

  v8f c0 = {}; v8f c1 = {}; v8f c2 = {}; v8f c3 = {};

#pragma unroll
  for (int kk = 0; kk < KPAD; kk += 4) {
    int k0 = kk + half * 2;
    v2f a = *(const v2f*)(Arow + k0);          // global_load_b64, 8B aligned
    const v2f* wp = &sW2[(k0 >> 1) * 64 + m];  // contiguous pair -> ds_load_b64
    v2f b0 = wp[0];
    v2f b1 = wp[16];
    v2f b2 = wp[32];
    v2f b3 = wp[48];
    c0 = __builtin_amdgcn_wmma_f32_16x16x4_f32(false, a, false, b0, (short)0, c0, false, false);
    c1 = __builtin_amdgcn_wmma_f32_16x16x4_f32(false, a, false, b1, (short)0, c1, false, false);
    c2 = __builtin_amdgcn_wmma_f32_16x16x4_f32(false, a, false, b2, (short)0, c2, false, false);
    c3 = __builtin_amdgcn_wmma_f32_16x16x4_f32(false, a, false, b3, (short)0, c3, false, false);
  }

  // C/D layout: VGPR r -> lanes 0-15: M=r, lanes 16-31: M=r+8; N = lane&15
  int nidx  = lane & 15;
  int r0    = rowBase + half * 8;
  float* o  = out + (long long)r0 * 64 + nidx;

  if (rowBase + 16 <= n) {
    // full tile: uniform scalar branch, stores with immediate offsets only
#pragma unroll
    for (int r = 0; r < 8; ++r) {
      o[r * 64 +  0] = c0[r];
      o[r * 64 + 16] = c1[r];
      o[r * 64 + 32] = c2[r];
      o[r * 64 + 48] = c3[r];
    }
  } else {
    // tail tile: per-row guard
#pragma unroll
    for (int r = 0; r < 8; ++r) {
      if (r0 + r < n) {
        o[r * 64 +  0] = c0[r];
        o[r * 64 + 16] = c1[r];
        o[r * 64 + 32] = c2[r];
        o[r * 64 + 48] = c3[r];
      }
    }
  }
}

// ---------------- edge scatter: agg[dst] += t[src] * dinv[src]*dinv[dst] ----------------
// one wave32 per edge, 2 columns per lane: coalesced b64 gather + L2-resident f32 atomics

__global__ void k_edge_scatter(const int* __restrict__ src, const int* __restrict__ dst,
                               const float* __restrict__ dinv, const float* __restrict__ t,
                               float* __restrict__ agg, unsigned E) {
  unsigned g = blockIdx.x * blockDim.x + threadIdx.x;
  unsigned e = g >> 5;
  int c2 = (int)(g & 31u) * 2;
  if (e >= E) return;
  int s = src[e], d = dst[e];
  float coef = dinv[s] * dinv[d];
  v2f v = *(const v2f*)(t + s * 64 + c2);
  float* ap = agg + d * 64 + c2;
  atomicAdd(ap,     v.x * coef);
  atomicAdd(ap + 1, v.y * coef);
}

// ---------------- fused self-loop + bias + BatchNorm(eval) + ReLU ----------------

__global__ void k_finalize(const float* __restrict__ agg, const float* __restrict__ t,
                           const float* __restrict__ dinv,
                           const float* __restrict__ bias, const float* __restrict__ gam,
                           const float* __restrict__ bet, const float* __restrict__ mu,
                           const float* __restrict__ var,
                           float* __restrict__ hout, unsigned n) {
  unsigned idx = blockIdx.x * blockDim.x + threadIdx.x;
  if (idx >= n * 64u) return;
  unsigned i = idx >> 6;
  int c = (int)(idx & 63u);
  float di = dinv[i];
  float conv = agg[idx] + t[idx] * (di * di) + bias[c];
  float y = (conv - mu[c]) * rsqrtf(var[c] + EPSBN) * gam[c] + bet[c];
  hout[idx] = fmaxf(y, 0.0f);
}

// ---------------- global mean pool (scatter phase), 2 columns per lane ----------------

__global__ void k_pool(const float* __restrict__ h, const int* __restrict__ batch,
                       float* __restrict__ pooled, float* __restrict__ counts, unsigned n) {
  unsigned g = blockIdx.x * blockDim.x + threadIdx.x;
  unsigned i = g >> 5;
  int c2 = (int)(g & 31u) * 2;
  if (i >= n) return;
  int b = batch[i];
  v2f v = *(const v2f*)(h + i * 64 + c2);
  atomicAdd(&pooled[b * 64 + c2],     v.x);
  atomicAdd(&pooled[b * 64 + c2 + 1], v.y);
  if (c2 == 0) atomicAdd(&counts[b], 1.0f);
}

// ---------------- MLP head: relu(pooled/cnt @ fw1 + fb1) @ fw2 + fb2 ----------------

__global__ void k_mlp(const float* __restrict__ pooled, const float* __restrict__ counts,
                      const float* __restrict__ fw1, const float* __restrict__ fb1,
                      const float* __restrict__ fw2, const float* __restrict__ fb2,
                      float* __restrict__ out) {
  int b = threadIdx.x;   // blockDim.x == B
  float inv = 1.0f / fmaxf(counts[b], 1.0f);
  float p[64];
#pragma unroll
  for (int i = 0; i < 64; ++i) p[i] = pooled[b * 64 + i] * inv;
  float acc0 = fb2[0], acc1 = fb2[1], acc2 = fb2[2];
  for (int j = 0; j < 32; ++j) {
    float hj = fb1[j];
#pragma unroll
    for (int i = 0; i < 64; ++i) hj = fmaf(p[i], fw1[i * 32 + j], hj);
    hj = fmaxf(hj, 0.0f);
    acc0 = fmaf(hj, fw2[j * 3 + 0], acc0);
    acc1 = fmaf(hj, fw2[j * 3 + 1], acc1);
    acc2 = fmaf(hj, fw2[j * 3 + 2], acc2);
  }
  out[b * 3 + 0] = acc0;
  out[b * 3 + 1] = acc1;
  out[b * 3 + 2] = acc2;
}

// ---------------- launcher ----------------

static inline unsigned gdiv(long long a, long long b) { return (unsigned)((a + b - 1) / b); }

extern "C" void kernel_launch(void* const* d_in, const int* in_sizes, int n_in,
                              void* d_out, int out_size, void* d_ws, size_t ws_size,
                              hipStream_t stream) {
  const float* x     = (const float*)d_in[0];
  const int*   ei    = (const int*)d_in[1];
  const int*   batch = (const int*)d_in[2];
  const float* W[3]  = {(const float*)d_in[3],  (const float*)d_in[9],  (const float*)d_in[15]};
  const float* bb[3] = {(const float*)d_in[4],  (const float*)d_in[10], (const float*)d_in[16]};
  const float* gg[3] = {(const float*)d_in[5],  (const float*)d_in[11], (const float*)d_in[17]};
  const float* be[3] = {(const float*)d_in[6],  (const float*)d_in[12], (const float*)d_in[18]};
  const float* mm[3] = {(const float*)d_in[7],  (const float*)d_in[13], (const float*)d_in[19]};
  const float* vv[3] = {(const float*)d_in[8],  (const float*)d_in[14], (const float*)d_in[20]};
  const float* fw1 = (const float*)d_in[21];
  const float* fb1 = (const float*)d_in[22];
  const float* fw2 = (const float*)d_in[23];
  const float* fb2 = (const float*)d_in[24];

  const long long N = in_sizes[2];          // batch array has one entry per node
  const long long E = in_sizes[1] / 2;      // edge_index is [2, E]
  const int F_IN = (int)(in_sizes[0] / N);
  const int KPAD1 = 12;                     // F_IN=10 padded to multiple of 4
  const int B = 128;

  // workspace (floats): xpad[N*12] | dinv[N] | t[N*64] | agg[N*64] | h[N*64] | pooled[B*64] | counts[B]
  float* ws     = (float*)d_ws;
  float* xpad   = ws;
  float* dinv   = xpad + N * KPAD1;
  float* t      = dinv + N;
  float* agg    = t + N * 64;
  float* h      = agg + N * 64;
  float* pooled = h + N * 64;
  float* counts = pooled + (long long)B * 64;

  const int* srcIdx = ei;
  const int* dstIdx = ei + E;

  // degrees with implicit self-loop: deg = 1 + scatter(1, dst); dinv = rsqrt(deg)
  k_fill<<<gdiv(N, 256), 256, 0, stream>>>(dinv, 1.0f, (unsigned)N);
  k_edge_deg<<<gdiv(E, 256), 256, 0, stream>>>(dstIdx, dinv, (unsigned)E);
  k_rsqrt_inplace<<<gdiv(N, 256), 256, 0, stream>>>(dinv, (unsigned)N);

  // zero-pad x once so all GEMMs run guard-free
  k_pad<<<gdiv(N * KPAD1, 256), 256, 0, stream>>>(x, xpad, (unsigned)N, F_IN, KPAD1);

  for (int l = 0; l < 3; ++l) {
    if (l == 0)
      k_gemm_wmma<KPAD1><<<gdiv(N, 128), 256, 0, stream>>>(xpad, W[0], t, (int)N, F_IN);
    else
      k_gemm_wmma<64><<<gdiv(N, 128), 256, 0, stream>>>(h, W[l], t, (int)N, 64);

    k_fill<<<gdiv(N * 64, 256), 256, 0, stream>>>(agg, 0.0f, (unsigned)(N * 64));
    k_edge_scatter<<<gdiv(E * 32, 256), 256, 0, stream>>>(srcIdx, dstIdx, dinv, t, agg,
                                                          (unsigned)E);
    k_finalize<<<gdiv(N * 64, 256), 256, 0, stream>>>(agg, t, dinv, bb[l], gg[l], be[l],
                                                      mm[l], vv[l], h, (unsigned)N);
  }

  // mean pool + head (pooled and counts are contiguous -> one zero pass)
  k_fill<<<gdiv((long long)B * 64 + B, 256), 256, 0, stream>>>(pooled, 0.0f,
                                                               (unsigned)(B * 64 + B));
  k_pool<<<gdiv(N * 32, 256), 256, 0, stream>>>(h, batch, pooled, counts, (unsigned)N);
  k_mlp<<<1, B, 0, stream>>>(pooled, counts, fw1, fb1, fw2, fb2, (float*)d_out);
}